// ViterbiLoss_21363167330842
// MI455X (gfx1250) — compile-verified
//
#include <hip/hip_runtime.h>
#include <hip/hip_bf16.h>
#include <stdint.h>

#define Bsz 32
#define Tmax 512
#define Ktags 48
#define START_TAG 46
#define STOP_TAG 47
#define TILE (Ktags * Ktags)        // 2304 floats
#define PIPE 6                      // 6 * 9216 B = 55296 B LDS tile pipeline
#define NTHREADS 96                 // 3 wave32 -> 3 x 16-row WMMA blocks covers j=0..47

typedef __attribute__((ext_vector_type(2))) float v2f;
typedef __attribute__((ext_vector_type(8))) float v8f;

// Each thread issues 6 async b128 loads; a 96-thread block covers one
// 9216-byte 48x48 f32 tile. Per wave this is exactly 6 ASYNCcnt increments.
__device__ __forceinline__ void async_tile_part(uint32_t lds_byte, const float* g, int tid) {
  uint32_t l = lds_byte + (uint32_t)tid * 16u;
  const char* p = (const char*)g + (size_t)tid * 16u;
#pragma unroll
  for (int k = 0; k < 6; ++k) {
    asm volatile("global_load_async_to_lds_b128 %0, %1, off"
                 :: "v"(l + (uint32_t)k * 1536u), "v"(p + k * 1536)
                 : "memory");
  }
}

__global__ __launch_bounds__(NTHREADS)
void crf_forward_kernel(const float* __restrict__ feats,
                        const int* __restrict__ targets,
                        const int* __restrict__ lengths,
                        float* __restrict__ acc /* acc[0]=paths, acc[1]=gold */) {
  __shared__ float tiles[PIPE][TILE];
  __shared__ float carr[Ktags];
  __shared__ float wexp[Ktags];

  const int b   = blockIdx.x;
  const int tid = threadIdx.x;
  const int len = lengths[b];                       // uniform per block
  const float* fb = feats + (size_t)b * Tmax * TILE;

  // ---- prime the pipeline: steps 0..PIPE-1 (clamped; dummies never read) ----
#pragma unroll
  for (int s = 0; s < PIPE; ++s) {
    const int ts = (s < len) ? s : (len - 1);
    async_tile_part((uint32_t)(uintptr_t)&tiles[s][0], fb + (size_t)ts * TILE, tid);
  }
  // outstanding beyond tile 0 = (PIPE-1) tiles * 6 instr = 30
  asm volatile("s_wait_asynccnt 30" ::: "memory");
  __syncthreads();

  float gold = 0.0f;
  if (tid == 0)     gold = tiles[0][targets[(size_t)b * Tmax + 0]];
  if (tid < Ktags)  carr[tid] = tiles[0][START_TAG * Ktags + tid];  // init row

  const int   lane  = tid & 31;
  const int   m     = lane & 15;          // WMMA row within 16-block
  const int   khalf = lane >> 4;          // selects K pair within chunk
  const int   jb    = (tid >> 5) * 16;    // wave id -> output j-block
  const float bmask = (m == 0) ? 1.0f : 0.0f;  // only column N=0 of B is live

  for (int t = 1; t < len; ++t) {
    // in-order async completion: <= (PIPE-2) tiles beyond tile t outstanding
    asm volatile("s_wait_asynccnt 24" ::: "memory");
    __syncthreads();  // tile t visible to all waves; slot (t-1)%PIPE fully consumed

    {   // refill the just-freed slot with tile t+PIPE-1 (clamped dummy past len)
      const int ss = t + PIPE - 1;
      const int ts = (ss < len) ? ss : (len - 1);
      async_tile_part((uint32_t)(uintptr_t)&tiles[(t - 1) % PIPE][0],
                      fb + (size_t)ts * TILE, tid);
    }

    const float* tile = tiles[t % PIPE];
    if (tid == 0) gold += tile[targets[(size_t)b * Tmax + t]];

    // mc = max(carry): branchless first fold (idempotent), then wave32 shuffle
    float x = fmaxf(carr[lane], carr[32 + (lane & 15)]);
#pragma unroll
    for (int off = 16; off > 0; off >>= 1)
      x = fmaxf(x, __shfl_xor(x, off, 32));
    const float mc = x;
    if (tid < Ktags) wexp[tid] = __expf(carr[tid] - mc);
    __syncthreads();  // wexp ready

    // S[j] = sum_i exp(tile[i*48+j]) * wexp[i]  via v_wmma_f32_16x16x4_f32
    // A(16x4): lane holds A[m, 2*khalf + {0,1}] ; B(4x16): column N=0 = w-chunk.
    // B built from unconditional broadcast LDS reads * bmask (no exec toggling).
    v8f s8 = {0.f, 0.f, 0.f, 0.f, 0.f, 0.f, 0.f, 0.f};
#pragma unroll
    for (int c = 0; c < 12; ++c) {
      const int i0 = 4 * c + 2 * khalf;
      v2f a, bb;
      a.x  = __expf(tile[i0       * Ktags + jb + m]);
      a.y  = __expf(tile[(i0 + 1) * Ktags + jb + m]);
      bb.x = wexp[i0]     * bmask;
      bb.y = wexp[i0 + 1] * bmask;
      s8 = __builtin_amdgcn_wmma_f32_16x16x4_f32(false, a, false, bb,
                                                 (short)0, s8, false, false);
    }
    // column N=0 of D: lane 0 -> M=0..7, lane 16 -> M=8..15
    if (m == 0) {
#pragma unroll
      for (int r = 0; r < 8; ++r)
        carr[jb + khalf * 8 + r] = mc + __logf(s8[r]);
    }
    // next loop-top barrier orders carr/wexp/tile reuse
  }

  __syncthreads();
  if (tid == 0) {
    atomicAdd(&acc[1], gold);
    atomicAdd(&acc[0], carr[STOP_TAG]);
  }
}

__global__ void init_acc_kernel(float* acc) {
  if (threadIdx.x < 2) acc[threadIdx.x] = 0.0f;
}

__global__ void finalize_kernel(const float* acc, float* out) {
  if (threadIdx.x == 0) out[0] = (acc[0] - acc[1]) / (float)Bsz;
}

extern "C" void kernel_launch(void* const* d_in, const int* in_sizes, int n_in,
                              void* d_out, int out_size, void* d_ws, size_t ws_size,
                              hipStream_t stream) {
  const float* feats   = (const float*)d_in[0];
  const int*   targets = (const int*)d_in[1];
  const int*   lengths = (const int*)d_in[2];
  float* acc = (float*)d_ws;   // [0]=all-paths score, [1]=gold score

  init_acc_kernel<<<1, 32, 0, stream>>>(acc);
  crf_forward_kernel<<<Bsz, NTHREADS, 0, stream>>>(feats, targets, lengths, acc);
  finalize_kernel<<<1, 32, 0, stream>>>(acc, (float*)d_out);
}